// EVLAGNNProcessor_6992206758398
// MI455X (gfx1250) — compile-verified
//
#include <hip/hip_runtime.h>

#define N_NODES 50000
#define N_EDGES 800000
#define IN_CH   5
#define HID_CH  64
#define OUT_CH  32

typedef float v2f __attribute__((ext_vector_type(2)));
typedef float v8f __attribute__((ext_vector_type(8)));

// Native non-returning fp32 atomic add (global_atomic_add_f32), device scope,
// relaxed ordering. Avoids any CAS-loop lowering of plain atomicAdd.
__device__ __forceinline__ void atomic_add_f32(float* p, float v) {
    __hip_atomic_fetch_add(p, v, __ATOMIC_RELAXED, __HIP_MEMORY_SCOPE_AGENT);
}

// ---------------- utility: zero fill ----------------
__global__ void zero_kernel(float* __restrict__ p, int n) {
    int i = blockIdx.x * blockDim.x + threadIdx.x;
    if (i < n) p[i] = 0.0f;
}

// ---------------- degree count ----------------
__global__ void deg_kernel(const long long* __restrict__ ei, float* __restrict__ deg) {
    int e = blockIdx.x * blockDim.x + threadIdx.x;
    if (e < N_EDGES) {
        int d = (int)ei[N_EDGES + e];
        atomic_add_f32(&deg[d], 1.0f);
    }
}

__global__ void invdeg_kernel(float* __restrict__ deg) {
    int i = blockIdx.x * blockDim.x + threadIdx.x;
    if (i < N_NODES) deg[i] = 1.0f / fmaxf(deg[i], 1.0f);
}

// ---------------- scatter-add: agg[dst][c] += feat[src][c] ----------------
template <int C>
__global__ void scatter_kernel(const long long* __restrict__ ei,
                               const float* __restrict__ feat,
                               float* __restrict__ agg) {
    unsigned tid = blockIdx.x * blockDim.x + threadIdx.x;
    if (tid < (unsigned)N_EDGES * C) {
        unsigned e = tid / C;
        unsigned c = tid - e * C;
        int s = (int)ei[e];
        int d = (int)ei[N_EDGES + e];
        atomic_add_f32(&agg[(unsigned)d * C + c], feat[(unsigned)s * C + c]);
    }
}

// ---------------- layer 1 (K=5): scalar fused GEMM ----------------
__global__ void gemm1_kernel(const float* __restrict__ x,
                             const float* __restrict__ agg,
                             const float* __restrict__ inv_deg,
                             const float* __restrict__ W1l,
                             const float* __restrict__ b1,
                             const float* __restrict__ W1r,
                             float* __restrict__ h1) {
    int tid = blockIdx.x * blockDim.x + threadIdx.x;
    if (tid >= N_NODES * HID_CH) return;
    int n = tid >> 6;        // / HID_CH
    int o = tid & 63;        // % HID_CH
    float s = inv_deg[n];
    float acc = b1[o];
#pragma unroll
    for (int k = 0; k < IN_CH; ++k) {
        acc += agg[n * IN_CH + k] * s * W1l[o * IN_CH + k]
             + x[n * IN_CH + k] * W1r[o * IN_CH + k];
    }
    h1[tid] = fmaxf(acc, 0.0f);
}

// ---------------- layers 2/3: WMMA fused dual-GEMM ----------------
// out[m][n] = act( (agg[m]*inv_deg[m]) @ Wl^T + bl + xin[m] @ Wr^T )
// One wave per 16x16 output tile; K = HID_CH = 64 -> 32 v_wmma_f32_16x16x4_f32.
template <int OUTC, bool RELU>
__global__ __launch_bounds__(256)
void sage_gemm_wmma(const float* __restrict__ agg,
                    const float* __restrict__ inv_deg,
                    const float* __restrict__ xin,
                    const float* __restrict__ Wl,   // [OUTC][HID_CH] row-major
                    const float* __restrict__ bl,   // [OUTC]
                    const float* __restrict__ Wr,   // [OUTC][HID_CH] row-major
                    float* __restrict__ out) {
    constexpr int K = HID_CH;
    constexpr int NT = OUTC / 16;                  // output-column tiles
    const int total_waves = (N_NODES / 16) * NT;   // 3125 * NT

    int wave = (blockIdx.x * blockDim.x + threadIdx.x) >> 5;
    if (wave >= total_waves) return;               // wave-uniform: EXEC stays all-1s

    const int m0 = (wave / NT) * 16;
    const int n0 = (wave % NT) * 16;
    const int lane  = threadIdx.x & 31;
    const int lrow  = lane & 15;   // A row / B-D column within tile
    const int khalf = lane >> 4;   // which K pair this half-wave holds

    const int m = m0 + lrow;
    const float s = inv_deg[m];

    const float* aRow  = agg + (unsigned)m * K + 2 * khalf;
    const float* xRow  = xin + (unsigned)m * K + 2 * khalf;
    const float* wlRow = Wl + (unsigned)(n0 + lrow) * K + 2 * khalf;
    const float* wrRow = Wr + (unsigned)(n0 + lrow) * K + 2 * khalf;

    v8f c = {0.f, 0.f, 0.f, 0.f, 0.f, 0.f, 0.f, 0.f};

#pragma unroll
    for (int k0 = 0; k0 < K; k0 += 4) {
        // neighbor-mean path: A = mean tile (scale folded into A load)
        v2f a = *(const v2f*)(aRow + k0);
        a.x *= s; a.y *= s;
        v2f bL = *(const v2f*)(wlRow + k0);
        c = __builtin_amdgcn_wmma_f32_16x16x4_f32(false, a, false, bL,
                                                  (short)0, c, false, false);
        // root path: A = x tile
        v2f ax = *(const v2f*)(xRow + k0);
        v2f bR = *(const v2f*)(wrRow + k0);
        c = __builtin_amdgcn_wmma_f32_16x16x4_f32(false, ax, false, bR,
                                                  (short)0, c, false, false);
    }

    const float bias = bl[n0 + lrow];
#pragma unroll
    for (int v = 0; v < 8; ++v) {
        int mm = m0 + v + 8 * khalf;
        float val = c[v] + bias;
        if (RELU) val = fmaxf(val, 0.0f);
        out[(unsigned)mm * OUTC + n0 + lrow] = val;
    }
}

// ---------------- launcher ----------------
extern "C" void kernel_launch(void* const* d_in, const int* in_sizes, int n_in,
                              void* d_out, int out_size, void* d_ws, size_t ws_size,
                              hipStream_t stream) {
    const float*     x   = (const float*)d_in[0];
    const long long* ei  = (const long long*)d_in[1];   // int64 [2][E]
    const float*     W1l = (const float*)d_in[2];
    const float*     b1  = (const float*)d_in[3];
    const float*     W1r = (const float*)d_in[4];
    const float*     W2l = (const float*)d_in[5];
    const float*     b2  = (const float*)d_in[6];
    const float*     W2r = (const float*)d_in[7];
    const float*     W3l = (const float*)d_in[8];
    const float*     b3  = (const float*)d_in[9];
    const float*     W3r = (const float*)d_in[10];
    float* out = (float*)d_out;

    // workspace layout (floats)
    float* ws      = (float*)d_ws;
    float* inv_deg = ws;                          // N (deg first, then inverted in place)
    float* agg     = inv_deg + N_NODES;           // N * 64 (reused each layer)
    float* h1      = agg + (size_t)N_NODES * HID_CH;  // N * 64
    float* h2      = h1  + (size_t)N_NODES * HID_CH;  // N * 64

    const int B = 256;
    auto cdiv = [](long long a, long long b) { return (int)((a + b - 1) / b); };

    // degree -> inv_deg
    zero_kernel<<<cdiv(N_NODES, B), B, 0, stream>>>(inv_deg, N_NODES);
    deg_kernel<<<cdiv(N_EDGES, B), B, 0, stream>>>(ei, inv_deg);
    invdeg_kernel<<<cdiv(N_NODES, B), B, 0, stream>>>(inv_deg);

    // ---- layer 1 (IN 5 -> HID 64), scalar ----
    zero_kernel<<<cdiv((long long)N_NODES * IN_CH, B), B, 0, stream>>>(agg, N_NODES * IN_CH);
    scatter_kernel<IN_CH><<<cdiv((long long)N_EDGES * IN_CH, B), B, 0, stream>>>(ei, x, agg);
    gemm1_kernel<<<cdiv((long long)N_NODES * HID_CH, B), B, 0, stream>>>(
        x, agg, inv_deg, W1l, b1, W1r, h1);

    // ---- layer 2 (HID 64 -> HID 64), WMMA ----
    zero_kernel<<<cdiv((long long)N_NODES * HID_CH, B), B, 0, stream>>>(agg, N_NODES * HID_CH);
    scatter_kernel<HID_CH><<<cdiv((long long)N_EDGES * HID_CH, B), B, 0, stream>>>(ei, h1, agg);
    {
        const int waves = (N_NODES / 16) * (HID_CH / 16);   // 12500
        sage_gemm_wmma<HID_CH, true><<<cdiv((long long)waves * 32, B), B, 0, stream>>>(
            agg, inv_deg, h1, W2l, b2, W2r, h2);
    }

    // ---- layer 3 (HID 64 -> OUT 32), WMMA, no relu, writes d_out ----
    zero_kernel<<<cdiv((long long)N_NODES * HID_CH, B), B, 0, stream>>>(agg, N_NODES * HID_CH);
    scatter_kernel<HID_CH><<<cdiv((long long)N_EDGES * HID_CH, B), B, 0, stream>>>(ei, h2, agg);
    {
        const int waves = (N_NODES / 16) * (OUT_CH / 16);   // 6250
        sage_gemm_wmma<OUT_CH, false><<<cdiv((long long)waves * 32, B), B, 0, stream>>>(
            agg, inv_deg, h2, W3l, b3, W3r, out);
    }
}